// CKA_Loss_8899172237725
// MI455X (gfx1250) — compile-verified
//
#include <hip/hip_runtime.h>
#include <math.h>

// ---------------------------------------------------------------------------
// CKA loss, reformulated in feature space:
//   HSIC_ij = ||X_i^T X_j||_F^2 - d_i.d_j - (2/N) w_i.w_j + (1/N^2) T_i T_j
// with d_i[n]=||x_i[n]||^2, w_i[n]=x_i[n].s_i - d_i[n], s_i=colsum(X_i),
// T_i = ||s_i||^2 - sum_n d_i[n].
// Dominant cost: syrk G = Y^T Y (3072x3072x4096 fp32) -> fp32 WMMA 16x16x4.
// LDS k-pair-interleaved layout: every WMMA operand is ONE aligned ds_load_b64
// (no VGPR repacking movs); pair-row stride 288 floats keeps the two
// half-waves of a fragment load on disjoint bank sets (288 mod 64 == 32).
// ---------------------------------------------------------------------------

typedef float v2f __attribute__((ext_vector_type(2)));
typedef float v8f __attribute__((ext_vector_type(8)));

#define M_LAYERS 6
#define N_SAMP   4096
#define D_FEAT   512
#define P_TOT    (M_LAYERS * D_FEAT)   // 3072
#define TILE     128
#define NTILES   (P_TOT / TILE)        // 24
#define NPAIRT   (NTILES * (NTILES + 1) / 2)  // 300 upper-triangle tiles
#define KC       32                    // k-chunk staged in LDS
#define KPAIRS   (KC / 2)              // 16 interleaved k-pair rows
#define PSTRIDE  288                   // floats per k-pair row: 2*128 + 32 pad

__device__ __forceinline__ float waveReduceSum(float v) {
#pragma unroll
  for (int off = 16; off > 0; off >>= 1) v += __shfl_xor(v, off, 32);
  return v;
}

__device__ __forceinline__ float blockReduce(float v, float* red, int tid) {
  red[tid] = v;
  __syncthreads();
#pragma unroll
  for (int st = 128; st > 0; st >>= 1) {
    if (tid < st) red[tid] += red[tid + st];
    __syncthreads();
  }
  float r = red[0];
  __syncthreads();
  return r;
}

// -------------------------------------------------------------------- s_i ---
__global__ __launch_bounds__(256) void colsum_kernel(const float* __restrict__ X,
                                                     float* __restrict__ s) {
  const int i = blockIdx.x >> 1;
  const int d = ((blockIdx.x & 1) << 8) + threadIdx.x;
  const float* p = X + (size_t)i * N_SAMP * D_FEAT + d;
  float a0 = 0.f, a1 = 0.f, a2 = 0.f, a3 = 0.f;
#pragma unroll 2
  for (int n = 0; n < N_SAMP; n += 4) {
    a0 += p[(size_t)(n + 0) * D_FEAT];
    a1 += p[(size_t)(n + 1) * D_FEAT];
    a2 += p[(size_t)(n + 2) * D_FEAT];
    a3 += p[(size_t)(n + 3) * D_FEAT];
  }
  s[i * D_FEAT + d] = (a0 + a1) + (a2 + a3);
}

// ------------------------------------------------------------- d_i, v_i -----
__global__ __launch_bounds__(256) void rowstats_kernel(const float* __restrict__ X,
                                                       const float* __restrict__ s,
                                                       float* __restrict__ dv,
                                                       float* __restrict__ vv) {
  const int r    = blockIdx.x * 8 + (threadIdx.x >> 5);  // one wave per row
  const int lane = threadIdx.x & 31;
  const int i    = r >> 12;  // r / N_SAMP
  const float4* xr = (const float4*)(X + (size_t)r * D_FEAT);
  const float4* sr = (const float4*)(s + i * D_FEAT);
  float ad = 0.f, av = 0.f;
#pragma unroll
  for (int t = 0; t < 4; ++t) {
    float4 x  = xr[lane + 32 * t];
    float4 sv = sr[lane + 32 * t];
    ad += x.x * x.x + x.y * x.y + x.z * x.z + x.w * x.w;
    av += x.x * sv.x + x.y * sv.y + x.z * sv.z + x.w * sv.w;
  }
  ad = waveReduceSum(ad);
  av = waveReduceSum(av);
  if (lane == 0) { dv[r] = ad; vv[r] = av; }
}

// ------------------------------- syrk tile -> Frobenius partial (fp32 WMMA) -
__global__ __launch_bounds__(256) void gram_fro_kernel(const float* __restrict__ X,
                                                       float* __restrict__ partials) {
  __shared__ __align__(16) float As[KPAIRS * PSTRIDE];
  __shared__ __align__(16) float Bs[KPAIRS * PSTRIDE];
  __shared__ float wred[8];

  // linear block id -> (tp, tq) with tp <= tq
  int rem = blockIdx.x;
  int tp  = 0;
  while (rem >= (NTILES - tp)) { rem -= (NTILES - tp); ++tp; }
  const int tq = tp + rem;

  const int P0 = tp * TILE, Q0 = tq * TILE;
  const int ip = P0 >> 9, dp = P0 & 511;  // tiles never straddle a layer (512 % 128 == 0)
  const int iq = Q0 >> 9, dq = Q0 & 511;

  const int tid  = threadIdx.x;
  const int lane = tid & 31;
  const int wave = tid >> 5;
  const int wr   = wave & 3;              // 32-row band
  const int wc   = wave >> 2;             // 64-col band
  const int lm   = lane & 15;
  const int kh   = (lane >> 4) << 1;      // K offset 0 / 2 per half-wave (f32 A/B layout)

  const int d4 = tid & 31;                // float4 column for LDS fill
  const int kp = tid >> 5;                // base k-pair row for LDS fill (0..7)

  v8f acc[2][4];
#pragma unroll
  for (int fr = 0; fr < 2; ++fr)
#pragma unroll
    for (int fc = 0; fc < 4; ++fc)
#pragma unroll
      for (int e = 0; e < 8; ++e) acc[fr][fc][e] = 0.f;

  const float* baseA = X + (size_t)ip * N_SAMP * D_FEAT + dp + 4 * d4;
  const float* baseB = X + (size_t)iq * N_SAMP * D_FEAT + dq + 4 * d4;

  // per-lane fragment read base offsets (floats); lanes 16-31 sit one pair-row
  // (+PSTRIDE) ahead => +32 banks => conflict-free vs lanes 0-15.
  const int aoff = (kh >> 1) * PSTRIDE + 2 * (32 * wr + lm);
  const int boff = (kh >> 1) * PSTRIDE + 2 * (64 * wc + lm);

  for (int n0 = 0; n0 < N_SAMP; n0 += KC) {
    // fill: two consecutive k-rows per thread, written k-pair interleaved,
    // as two contiguous b128 stores per tile.
#pragma unroll
    for (int ss = 0; ss < 2; ++ss) {
      const int kpp = kp + 8 * ss;
      const int k   = 2 * kpp;
      float4 va0 = *(const float4*)(baseA + (size_t)(n0 + k) * D_FEAT);
      float4 va1 = *(const float4*)(baseA + (size_t)(n0 + k + 1) * D_FEAT);
      float4 vb0 = *(const float4*)(baseB + (size_t)(n0 + k) * D_FEAT);
      float4 vb1 = *(const float4*)(baseB + (size_t)(n0 + k + 1) * D_FEAT);
      float4* pa = (float4*)&As[kpp * PSTRIDE + 8 * d4];
      float4* pb = (float4*)&Bs[kpp * PSTRIDE + 8 * d4];
      pa[0] = make_float4(va0.x, va1.x, va0.y, va1.y);
      pa[1] = make_float4(va0.z, va1.z, va0.w, va1.w);
      pb[0] = make_float4(vb0.x, vb1.x, vb0.y, vb1.y);
      pb[1] = make_float4(vb0.z, vb1.z, vb0.w, vb1.w);
    }
    __syncthreads();

#pragma unroll
    for (int k0 = 0; k0 < KC; k0 += 4) {
      const float* Ab = As + (k0 >> 1) * PSTRIDE;
      const float* Bb = Bs + (k0 >> 1) * PSTRIDE;
      // each operand: one aligned ds_load_b64 into an even VGPR pair
      v2f a0 = *(const v2f*)(Ab + aoff);
      v2f a1 = *(const v2f*)(Ab + aoff + 32);
      v2f b0 = *(const v2f*)(Bb + boff);
      v2f b1 = *(const v2f*)(Bb + boff + 32);
      v2f b2 = *(const v2f*)(Bb + boff + 64);
      v2f b3 = *(const v2f*)(Bb + boff + 96);
      acc[0][0] = __builtin_amdgcn_wmma_f32_16x16x4_f32(false, a0, false, b0, (short)0, acc[0][0], false, false);
      acc[0][1] = __builtin_amdgcn_wmma_f32_16x16x4_f32(false, a0, false, b1, (short)0, acc[0][1], false, false);
      acc[0][2] = __builtin_amdgcn_wmma_f32_16x16x4_f32(false, a0, false, b2, (short)0, acc[0][2], false, false);
      acc[0][3] = __builtin_amdgcn_wmma_f32_16x16x4_f32(false, a0, false, b3, (short)0, acc[0][3], false, false);
      acc[1][0] = __builtin_amdgcn_wmma_f32_16x16x4_f32(false, a1, false, b0, (short)0, acc[1][0], false, false);
      acc[1][1] = __builtin_amdgcn_wmma_f32_16x16x4_f32(false, a1, false, b1, (short)0, acc[1][1], false, false);
      acc[1][2] = __builtin_amdgcn_wmma_f32_16x16x4_f32(false, a1, false, b2, (short)0, acc[1][2], false, false);
      acc[1][3] = __builtin_amdgcn_wmma_f32_16x16x4_f32(false, a1, false, b3, (short)0, acc[1][3], false, false);
    }
    __syncthreads();
  }

  // Frobenius sum of this 128x128 tile of G (every element appears once)
  float fs = 0.f;
#pragma unroll
  for (int fr = 0; fr < 2; ++fr)
#pragma unroll
    for (int fc = 0; fc < 4; ++fc)
#pragma unroll
      for (int e = 0; e < 8; ++e) { const float x = acc[fr][fc][e]; fs += x * x; }

  fs = waveReduceSum(fs);
  if (lane == 0) wred[wave] = fs;
  __syncthreads();
  if (tid == 0) {
    float t = 0.f;
#pragma unroll
    for (int w = 0; w < 8; ++w) t += wred[w];
    partials[blockIdx.x] = t;  // no atomics -> bitwise deterministic
  }
}

// -------------------------------------------------------------- finalize ----
__global__ __launch_bounds__(256) void finalize_kernel(const float* __restrict__ partials,
                                                       const float* __restrict__ s,
                                                       const float* __restrict__ dv,
                                                       const float* __restrict__ vv,
                                                       float* __restrict__ out) {
  __shared__ float red[256];
  __shared__ float dd[36];
  __shared__ float ww[36];
  __shared__ float T[6];
  const int tid = threadIdx.x;

  // pair reductions over N (uniform control flow around barriers)
  for (int p = 0; p < 21; ++p) {
    int i = 0, rem = p;
    while (rem >= (M_LAYERS - i)) { rem -= (M_LAYERS - i); ++i; }
    const int j = i + rem;
    float a1 = 0.f, a2 = 0.f;
    for (int n = tid; n < N_SAMP; n += 256) {
      const float di = dv[i * N_SAMP + n], dj = dv[j * N_SAMP + n];
      const float wi = vv[i * N_SAMP + n] - di;
      const float wj = vv[j * N_SAMP + n] - dj;
      a1 += di * dj;
      a2 += wi * wj;
    }
    const float r1 = blockReduce(a1, red, tid);
    const float r2 = blockReduce(a2, red, tid);
    if (tid == 0) { dd[i * 6 + j] = dd[j * 6 + i] = r1; ww[i * 6 + j] = ww[j * 6 + i] = r2; }
  }
  // T_i
  for (int i = 0; i < M_LAYERS; ++i) {
    float a = 0.f;
    for (int n = tid; n < N_SAMP; n += 256) a += dv[i * N_SAMP + n];
    const float dsum = blockReduce(a, red, tid);
    float b = 0.f;
    for (int d = tid; d < D_FEAT; d += 256) { const float x = s[i * D_FEAT + d]; b += x * x; }
    const float ssq = blockReduce(b, red, tid);
    if (tid == 0) T[i] = ssq - dsum;
  }
  __syncthreads();

  if (tid == 0) {
    float A[36];
    for (int e = 0; e < 36; ++e) A[e] = 0.f;
    int b = 0;
    for (int tp = 0; tp < NTILES; ++tp)
      for (int tq = tp; tq < NTILES; ++tq, ++b) {
        const float F = partials[b];
        const int i = tp >> 2, j = tq >> 2;   // 4 tiles per 512-wide layer block
        A[i * 6 + j] += F;
        if (tp != tq) A[j * 6 + i] += F;      // mirror tile (symmetry of G)
      }
    const float invN = 1.0f / (float)N_SAMP;
    float S[36];
    for (int i = 0; i < 6; ++i)
      for (int j = 0; j < 6; ++j) {
        const float h = A[i * 6 + j] - dd[i * 6 + j]
                      - 2.0f * invN * ww[i * 6 + j]
                      + invN * invN * T[i] * T[j];
        S[i * 6 + j] = fabsf(h);
      }
    float vis[36];
    for (int e = 0; e < 36; ++e) vis[e] = 0.f;
    for (int i = 0; i < 6; ++i) vis[i * 6 + i] = 1.0f;
    float l = 0.f;
    for (int i = 1; i < 6; ++i)
      for (int j = 0; j < i; ++j) {
        const float c = S[i * 6 + j] / sqrtf(S[i * 6 + i] * S[j * 6 + j] + 1e-6f);
        vis[i * 6 + j] = c;
        vis[j * 6 + i] = c;
        l += c;
      }
    for (int e = 0; e < 36; ++e) out[e] = vis[e];
    out[36] = l;
  }
}

// ---------------------------------------------------------------------------
extern "C" void kernel_launch(void* const* d_in, const int* in_sizes, int n_in,
                              void* d_out, int out_size, void* d_ws, size_t ws_size,
                              hipStream_t stream) {
  const float* X  = (const float*)d_in[0];  // [6, 4096, 512] fp32
  float* out = (float*)d_out;               // 36 (hsic_visual) + 1 (l)
  float* ws  = (float*)d_ws;                // ~206 KB used, fully rewritten each call
  float* s        = ws;                                 // 3072
  float* dvv      = s + M_LAYERS * D_FEAT;              // 24576
  float* vvv      = dvv + M_LAYERS * N_SAMP;            // 24576
  float* partials = vvv + M_LAYERS * N_SAMP;            // 300

  colsum_kernel  <<<M_LAYERS * 2, 256, 0, stream>>>(X, s);
  rowstats_kernel<<<(M_LAYERS * N_SAMP) / 8, 256, 0, stream>>>(X, s, dvv, vvv);
  gram_fro_kernel<<<NPAIRT, 256, 0, stream>>>(X, partials);
  finalize_kernel<<<1, 256, 0, stream>>>(partials, s, dvv, vvv, out);
}